// MMDLoss_38611755991384
// MI455X (gfx1250) — compile-verified
//
#include <hip/hip_runtime.h>
#include <hip/hip_bf16.h>

typedef float v2f __attribute__((ext_vector_type(2)));
typedef float v8f __attribute__((ext_vector_type(8)));

// ---------------------------------------------------------------------------
// Kernel 1: squared L2 norm of each 64-element row.
// One wave per row: each lane loads a float2 (cols 2*lane, 2*lane+1), squares,
// wave-reduces via shfl_xor (wave32).
// ---------------------------------------------------------------------------
__global__ void mmd_row_norms(const float* __restrict__ x,
                              const float* __restrict__ y,
                              float* __restrict__ nx,
                              float* __restrict__ ny,
                              int rows) {
  int wave = threadIdx.x >> 5;
  int lane = threadIdx.x & 31;
  int row  = blockIdx.x * (blockDim.x >> 5) + wave;

  const float* src;
  float* dst;
  if (row < rows) { src = x + (size_t)row * 64;          dst = nx + row; }
  else            { src = y + (size_t)(row - rows) * 64; dst = ny + (row - rows); }

  float2 v = *(const float2*)(src + 2 * lane);
  float s = v.x * v.x + v.y * v.y;
#pragma unroll
  for (int off = 16; off > 0; off >>= 1) s += __shfl_xor(s, off, 32);
  if (lane == 0) *dst = s;
}

// ---------------------------------------------------------------------------
// Kernel 2: one wave per 32x32 output block = 2x2 WMMA tiles.
//  - 2x2 register blocking: 4 float2 loads feed 4 v_wmma_f32_16x16x4_f32 per
//    K-step (8 FLOP/byte from L2), 4 independent accumulation chains.
//  - Software pipelined: loads for K-step k+1 are issued before the WMMAs of
//    step k, so each s_wait_loadcnt sits a full WMMA group after its loads.
//
// f32 WMMA layouts (g = lane/16, n16 = lane%16):
//   A(16x4): lane holds A[n16][2g], A[n16][2g+1]      -> contiguous float2
//   B(4x16): lane holds B[2g][n16], B[2g+1][n16]      -> contiguous float2
//   C(16x16): lane, VGPR r -> element (M = r + 8g, N = n16)
//
// Work decomposition per batch (side = N/32 = 64 blocks):
//   xy: full side*side blocks, weight +1
//   xx: upper triangle ti<=tj, weight -0.5 (diag) / -1.0 (off-diag, mirrored)
//   yy: same with y                      -> 33% fewer FLOPs vs full matrices
// ---------------------------------------------------------------------------
__global__ void mmd_energy_tiles(const float* __restrict__ X,
                                 const float* __restrict__ Y,
                                 const float* __restrict__ NX,
                                 const float* __restrict__ NY,
                                 float* __restrict__ partials,
                                 int side,       // N/32
                                 int nFull,      // side*side
                                 int nTri,       // side*(side+1)/2
                                 int N) {
  __shared__ float wsum[8];

  const int wave = threadIdx.x >> 5;
  const int lane = threadIdx.x & 31;
  const int t    = blockIdx.x * 8 + wave;        // wave-uniform

  const int perBatch = nFull + 2 * nTri;
  const int batch = t / perBatch;
  int u           = t - batch * perBatch;

  // Decode (kind, ti, tj, weight). All wave-uniform scalar work.
  int kind, ti, tj;
  float w;
  if (u < nFull) {
    kind = 0;
    ti = u / side;
    tj = u - ti * side;
    w = 1.0f;
  } else {
    u -= nFull;
    kind = (u < nTri) ? 1 : 2;
    if (kind == 2) u -= nTri;
    // triangular decode: u = sum_{r<ti}(side - r) + (tj - ti), ti <= tj
    int i = 0, rem = u;
    while (rem >= side - i) { rem -= side - i; ++i; }
    ti = i;
    tj = i + rem;
    w = (ti == tj) ? -0.5f : -1.0f;   // off-diag counts mirrored block too
  }

  const size_t boff = (size_t)batch * N * 64;
  const size_t noff = (size_t)batch * N;

  const float *A, *Bm, *nA, *nB;
  if (kind == 0)      { A = X + boff; Bm = Y + boff; nA = NX + noff; nB = NY + noff; }
  else if (kind == 1) { A = X + boff; Bm = X + boff; nA = NX + noff; nB = NX + noff; }
  else                { A = Y + boff; Bm = Y + boff; nA = NY + noff; nB = NY + noff; }

  const int n16 = lane & 15;
  const int g   = lane >> 4;
  const int m0  = ti * 32;
  const int n0  = tj * 32;

  const float* ar0 = A  + (size_t)(m0 + n16) * 64      + 2 * g;
  const float* ar1 = A  + (size_t)(m0 + 16 + n16) * 64 + 2 * g;
  const float* br0 = Bm + (size_t)(n0 + n16) * 64      + 2 * g;
  const float* br1 = Bm + (size_t)(n0 + 16 + n16) * 64 + 2 * g;

  // Prologue loads for K-step 0.
  float2 a0v = *(const float2*)(ar0);
  float2 a1v = *(const float2*)(ar1);
  float2 b0v = *(const float2*)(br0);
  float2 b1v = *(const float2*)(br1);

  v8f c00 = {}, c01 = {}, c10 = {}, c11 = {};
#pragma unroll
  for (int k = 0; k < 16; ++k) {
    // Issue next K-step's loads before consuming this step's data.
    float2 na0, na1, nb0, nb1;
    if (k < 15) {
      na0 = *(const float2*)(ar0 + 4 * (k + 1));
      na1 = *(const float2*)(ar1 + 4 * (k + 1));
      nb0 = *(const float2*)(br0 + 4 * (k + 1));
      nb1 = *(const float2*)(br1 + 4 * (k + 1));
    }
    v2f a0; a0.x = a0v.x; a0.y = a0v.y;
    v2f a1; a1.x = a1v.x; a1.y = a1v.y;
    v2f b0; b0.x = b0v.x; b0.y = b0v.y;
    v2f b1; b1.x = b1v.x; b1.y = b1v.y;
    c00 = __builtin_amdgcn_wmma_f32_16x16x4_f32(false, a0, false, b0, (short)0, c00, false, false);
    c01 = __builtin_amdgcn_wmma_f32_16x16x4_f32(false, a0, false, b1, (short)0, c01, false, false);
    c10 = __builtin_amdgcn_wmma_f32_16x16x4_f32(false, a1, false, b0, (short)0, c10, false, false);
    c11 = __builtin_amdgcn_wmma_f32_16x16x4_f32(false, a1, false, b1, (short)0, c11, false, false);
    if (k < 15) { a0v = na0; a1v = na1; b0v = nb0; b1v = nb1; }
  }

  // Epilogue: distance = sqrt(max(nx[m] + ny[n] - 2*G, 0)), accumulate.
  const float ny0 = nB[n0 + n16];
  const float ny1 = nB[n0 + 16 + n16];
  const float* nx0 = nA + m0 + 8 * g;
  const float* nx1 = nA + m0 + 16 + 8 * g;

  float acc = 0.0f;
#pragma unroll
  for (int rr = 0; rr < 8; ++rr) {
    float nxa = nx0[rr];
    float nxb = nx1[rr];
    float s00 = fmaxf(nxa + ny0 - 2.0f * c00[rr], 0.0f);
    float s01 = fmaxf(nxa + ny1 - 2.0f * c01[rr], 0.0f);
    float s10 = fmaxf(nxb + ny0 - 2.0f * c10[rr], 0.0f);
    float s11 = fmaxf(nxb + ny1 - 2.0f * c11[rr], 0.0f);
    acc += __builtin_amdgcn_sqrtf(s00);
    acc += __builtin_amdgcn_sqrtf(s01);
    acc += __builtin_amdgcn_sqrtf(s10);
    acc += __builtin_amdgcn_sqrtf(s11);
  }
  acc *= w;

#pragma unroll
  for (int off = 16; off > 0; off >>= 1) acc += __shfl_xor(acc, off, 32);
  if (lane == 0) wsum[wave] = acc;
  __syncthreads();
  if (threadIdx.x == 0) {
    float s = 0.0f;
#pragma unroll
    for (int i = 0; i < 8; ++i) s += wsum[i];
    partials[blockIdx.x] = s;
  }
}

// ---------------------------------------------------------------------------
// Kernel 3: deterministic fixed-order reduction of block partials -> scalar.
// ---------------------------------------------------------------------------
__global__ void mmd_final_reduce(const float* __restrict__ partials, int n,
                                 float* __restrict__ out, float scale) {
  __shared__ float sm[256];
  float s = 0.0f;
  for (int i = threadIdx.x; i < n; i += 256) s += partials[i];
  sm[threadIdx.x] = s;
  __syncthreads();
#pragma unroll
  for (int off = 128; off > 0; off >>= 1) {
    if ((int)threadIdx.x < off) sm[threadIdx.x] += sm[threadIdx.x + off];
    __syncthreads();
  }
  if (threadIdx.x == 0) out[0] = sm[0] * scale;
}

// ---------------------------------------------------------------------------
extern "C" void kernel_launch(void* const* d_in, const int* in_sizes, int n_in,
                              void* d_out, int out_size, void* d_ws, size_t ws_size,
                              hipStream_t stream) {
  const float* x = (const float*)d_in[0];
  const float* y = (const float*)d_in[1];

  const int D = 64;
  const int N = 2048;                 // downsample (reference setup)
  const int total  = in_sizes[0];     // 32768 * 64
  const int rows   = total / D;       // 32768
  const int nBatch = rows / N;        // 16

  float* ws       = (float*)d_ws;
  float* nx       = ws;               // rows floats
  float* ny       = ws + rows;        // rows floats
  float* partials = ws + 2 * rows;    // one float per tile-block

  // Kernel 1: 8 waves/block, one row per wave.
  const int normBlocks = (2 * rows) / 8;
  mmd_row_norms<<<normBlocks, 256, 0, stream>>>(x, y, nx, ny, rows);

  // Kernel 2: one 32x32 block per wave, 8 waves/block.
  const int side   = N / 32;                    // 64
  const int nFull  = side * side;               // 4096  (xy)
  const int nTri   = side * (side + 1) / 2;     // 2080  (xx or yy, upper tri)
  const int perBatch   = nFull + 2 * nTri;      // 8256
  const int totalWaves = nBatch * perBatch;     // 132096
  const int tileBlocks = totalWaves / 8;        // 16512
  mmd_energy_tiles<<<tileBlocks, 256, 0, stream>>>(x, y, nx, ny, partials,
                                                   side, nFull, nTri, N);

  // Kernel 3: scalar, scale = 1 / (B * N * N).
  const float scale = 1.0f / ((float)nBatch * (float)N * (float)N);
  mmd_final_reduce<<<1, 256, 0, stream>>>(partials, tileBlocks,
                                          (float*)d_out, scale);
}